// MixtureLayer_76768245448881
// MI455X (gfx1250) — compile-verified
//
#include <hip/hip_runtime.h>
#include <hip/hip_bf16.h>
#include <math.h>

// ---- problem constants (from reference) ----
#define BATCH   16384
#define INSIZE  1024
#define N_MIX   32
#define N_TGT   16
#define TRI     136            // 16*17/2
#define N_OUT   4896           // 32*(1+16+136)
#define OUT_ROW 8736           // 32 + 512 + 32*256
#define COL_TILES 306          // 4896/16

#define KCHUNK  64
#define NCHUNK  (INSIZE / KCHUNK)      // 16
#define ROWS_WG 64                     // 4 row tiles of 16 per wave
#define ALDS_STRIDE 68                 // 68 % 64 == 4 -> conflict free, 16B aligned

typedef float v2f __attribute__((ext_vector_type(2)));
typedef float v8f __attribute__((ext_vector_type(8)));
typedef int   v4i __attribute__((ext_vector_type(4)));

#if __has_builtin(__builtin_amdgcn_global_load_async_to_lds_b128)
#define HAVE_ASYNC_LDS 1
typedef __attribute__((address_space(1))) v4i gas_v4i;   // global 128b payload
typedef __attribute__((address_space(3))) v4i las_v4i;   // LDS 128b payload
#else
#define HAVE_ASYNC_LDS 0
#endif

__device__ __forceinline__ void wait_async_lds() {
#if __has_builtin(__builtin_amdgcn_s_wait_asynccnt)
    __builtin_amdgcn_s_wait_asynccnt(0);
#else
    asm volatile("s_wait_asynccnt 0x0" ::: "memory");
#endif
}

// -----------------------------------------------------------------------------
// Kernel 1: C[16384 x 4896] = A[16384 x 1024] * W[1024 x 4896] + b
// fp32 WMMA 16x16x4. 256 threads = 8 waves; each wave owns a 64x16 output
// strip (4 row tiles sharing each B fragment -> 4x fewer W loads per WMMA).
// A is staged with GLOBAL_LOAD_ASYNC_TO_LDS_B128 into a double-buffered LDS
// tile (2 x 64x68 f32) so the copy of chunk c+1 overlaps the WMMA loop on
// chunk c (ASYNCcnt + barrier handoff).
// Result stored into d_out[row*OUT_ROW + col] (front of each output row,
// consumed + overwritten by the epilogue kernel).
// -----------------------------------------------------------------------------
__global__ __launch_bounds__(256) void mixture_gemm_wmma(
    const float* __restrict__ A, const float* __restrict__ W,
    const float* __restrict__ bias, float* __restrict__ out)
{
    __shared__ float Alds[2][ROWS_WG * ALDS_STRIDE];   // 2 x 17.4 KB

    const int tid  = threadIdx.x;
    const int wave = tid >> 5;
    const int lane = tid & 31;
    const int half = lane >> 4;        // 0: lanes 0-15, 1: lanes 16-31
    const int l16  = lane & 15;

    const int rt = blockIdx.y;                    // 64-row tile
    const int ct = blockIdx.x * 8 + wave;         // 16-col tile
    const int ctc = (ct < COL_TILES) ? ct : (COL_TILES - 1);
    const int col = ctc * 16 + l16;

    const float* __restrict__ Arow = A + (size_t)rt * ROWS_WG * INSIZE;
    const float* __restrict__ Wc   = W + col;

    // ---- A-chunk staging: 64 rows x 64 floats = 1024 x b128, 4 per thread ----
    auto stage = [&](int kc, int buf) {
        #pragma unroll
        for (int t = 0; t < 4; ++t) {
            const int i  = tid + t * 256;
            const int r  = i >> 4;              // 16 b128 per row
            const int c4 = (i & 15) * 4;
            const float* g = Arow + (size_t)r * INSIZE + kc + c4;
            float* l = &Alds[buf][r * ALDS_STRIDE + c4];
#if HAVE_ASYNC_LDS
            __builtin_amdgcn_global_load_async_to_lds_b128(
                (gas_v4i*)g, (las_v4i*)l, 0, 0);
#else
            *(float4*)l = *(const float4*)g;
#endif
        }
    };

    v8f acc0 = {}, acc1 = {}, acc2 = {}, acc3 = {};

    stage(0, 0);
    wait_async_lds();
    __syncthreads();

    for (int c = 0; c < NCHUNK; ++c) {
        if (c + 1 < NCHUNK) stage((c + 1) * KCHUNK, (c + 1) & 1);

        const float* __restrict__ buf = &Alds[c & 1][0];
        const int kc = c * KCHUNK;

        #pragma unroll 4
        for (int kk = 0; kk < KCHUNK; kk += 4) {
            // B fragment (4x16): lane half selects K pair, one column per lane
            const int k = kc + kk + half * 2;
            v2f bfrag;
            bfrag.x = Wc[(size_t)k * N_OUT];
            bfrag.y = Wc[(size_t)(k + 1) * N_OUT];
            // A fragments (16x4) for the 4 row tiles
            const int ao = kk + half * 2;
            v2f a0 = *(const v2f*)&buf[(l16     ) * ALDS_STRIDE + ao];
            v2f a1 = *(const v2f*)&buf[(l16 + 16) * ALDS_STRIDE + ao];
            v2f a2 = *(const v2f*)&buf[(l16 + 32) * ALDS_STRIDE + ao];
            v2f a3 = *(const v2f*)&buf[(l16 + 48) * ALDS_STRIDE + ao];
            acc0 = __builtin_amdgcn_wmma_f32_16x16x4_f32(
                       false, a0, false, bfrag, (short)0, acc0, false, false);
            acc1 = __builtin_amdgcn_wmma_f32_16x16x4_f32(
                       false, a1, false, bfrag, (short)0, acc1, false, false);
            acc2 = __builtin_amdgcn_wmma_f32_16x16x4_f32(
                       false, a2, false, bfrag, (short)0, acc2, false, false);
            acc3 = __builtin_amdgcn_wmma_f32_16x16x4_f32(
                       false, a3, false, bfrag, (short)0, acc3, false, false);
        }

        wait_async_lds();     // chunk c+1 resident before next round
        __syncthreads();      // all waves done reading buf[c&1] before restage
    }

    if (ct < COL_TILES) {
        const float bv = bias[col];
        const size_t rbase = (size_t)rt * ROWS_WG;
        const v8f accs[4] = {acc0, acc1, acc2, acc3};
        #pragma unroll
        for (int t = 0; t < 4; ++t) {
            #pragma unroll
            for (int v = 0; v < 8; ++v) {
                const int m = t * 16 + v + half * 8;   // C layout: VGPR v -> M=v / v+8
                out[(rbase + m) * OUT_ROW + ct * 16 + l16] = accs[t][v] + bv;
            }
        }
    }
}

// -----------------------------------------------------------------------------
// Kernel 2: per-row epilogue. One wave (32 lanes) per batch row; lane = mixture.
//  - stage the 4896 GEMM outputs in LDS (then d_out row is safe to overwrite)
//  - softmax over 32 priors via wave32 shuffles
//  - mu transpose-gather
//  - sigma = L^T L + eps*I in packed lower-tri LDS (stride 33, conflict-free)
//  - in-place Cholesky, coalesced write-out with explicit upper-zero fill
// -----------------------------------------------------------------------------
__global__ __launch_bounds__(32) void mixture_epilogue(float* __restrict__ out)
{
    __shared__ float rowbuf[N_OUT];        // 19.6 KB
    __shared__ float ch[TRI * 33];         // 17.9 KB packed tri, per-mixture cols

    const int m = threadIdx.x;             // mixture id, 0..31
    float* __restrict__ o = out + (size_t)blockIdx.x * OUT_ROW;

    // stage GEMM result (front 4896 floats of the row) into LDS
    for (int i = m; i < N_OUT / 4; i += 32)
        ((float4*)rowbuf)[i] = ((const float4*)o)[i];
    __syncthreads();

    // ---- prior softmax (32 values, one per lane) ----
    const float x = rowbuf[m];
    float mx = x;
    #pragma unroll
    for (int off = 16; off; off >>= 1) mx = fmaxf(mx, __shfl_xor(mx, off, 32));
    const float e = __expf(x - mx);
    float s = e;
    #pragma unroll
    for (int off = 16; off; off >>= 1) s += __shfl_xor(s, off, 32);
    const float prior = e / s + 1e-9f;

    // ---- sigma = L^T L + eps*I (packed lower: sigma[i][k], k<=i) ----
    // L[j][i] = rowbuf[544 + (j*(j+1)/2 + i)*32 + m]   (tril row-major order)
    for (int i = 0; i < N_TGT; ++i) {
        for (int k = 0; k <= i; ++k) {
            float acc = (i == k) ? 1e-4f : 0.0f;
            for (int j = i; j < N_TGT; ++j) {           // L[j][i]=0 for j<i
                const int tj = (j * (j + 1)) >> 1;
                acc += rowbuf[544 + (tj + i) * 32 + m] *
                       rowbuf[544 + (tj + k) * 32 + m];
            }
            ch[(((i * (i + 1)) >> 1) + k) * 33 + m] = acc;
        }
    }

    // ---- in-place Cholesky on packed lower-tri ----
    for (int i = 0; i < N_TGT; ++i) {
        const int ti = (i * (i + 1)) >> 1;
        for (int k = 0; k < i; ++k) {
            const int tk = (k * (k + 1)) >> 1;
            float v = ch[(ti + k) * 33 + m];
            for (int j = 0; j < k; ++j)
                v -= ch[(ti + j) * 33 + m] * ch[(tk + j) * 33 + m];
            ch[(ti + k) * 33 + m] = v / ch[(tk + k) * 33 + m];
        }
        float d = ch[(ti + i) * 33 + m];
        for (int j = 0; j < i; ++j) {
            const float t = ch[(ti + j) * 33 + m];
            d -= t * t;
        }
        ch[(ti + i) * 33 + m] = sqrtf(fmaxf(d, 0.0f));
    }
    __syncthreads();

    // ---- writes ----
    o[m] = prior;                                        // prior[32]
    #pragma unroll
    for (int t = 0; t < N_TGT; ++t)                      // mu[b, m*16 + t]
        o[32 + m * N_TGT + t] = rowbuf[32 + t * 32 + m];

    // chol[b, m*256 + i*16 + k], cooperative coalesced writes, zero upper
    for (int mm = 0; mm < N_MIX; ++mm) {
        for (int idx = m; idx < 256; idx += 32) {
            const int i = idx >> 4;
            const int k = idx & 15;
            float v = 0.0f;
            if (k <= i) v = ch[(((i * (i + 1)) >> 1) + k) * 33 + mm];
            o[544 + mm * 256 + idx] = v;
        }
    }
}

extern "C" void kernel_launch(void* const* d_in, const int* in_sizes, int n_in,
                              void* d_out, int out_size, void* d_ws, size_t ws_size,
                              hipStream_t stream) {
    (void)in_sizes; (void)n_in; (void)d_ws; (void)ws_size; (void)out_size;
    const float* A    = (const float*)d_in[0];   // (16384, 1024)
    const float* W    = (const float*)d_in[1];   // (1024, 4896)
    const float* bias = (const float*)d_in[2];   // (4896,)
    float* out = (float*)d_out;                  // (16384, 8736)

    dim3 g1((COL_TILES + 7) / 8, BATCH / ROWS_WG);    // 39 x 256
    mixture_gemm_wmma<<<g1, 256, 0, stream>>>(A, W, bias, out);

    mixture_epilogue<<<BATCH, 32, 0, stream>>>(out);
}